// GraphConnectionEncoder_13211319402650
// MI455X (gfx1250) — compile-verified
//
#include <hip/hip_runtime.h>
#include <hip/hip_bf16.h>

typedef float v2f __attribute__((ext_vector_type(2)));
typedef float v8f __attribute__((ext_vector_type(8)));

#define DCOLS 128
#define KC 64
#define LDA 68    // padded row stride for A tile (16 x 64)
#define LDB 136   // padded row stride for B tile (64 x 128)

// ---- monotone float <-> uint key for atomic max on floats --------------------
__device__ __forceinline__ unsigned int fkey(float f) {
    unsigned int i = __float_as_uint(f);
    return (i & 0x80000000u) ? ~i : (i | 0x80000000u);
}
__device__ __forceinline__ float funkey(unsigned int k) {
    return (k & 0x80000000u) ? __uint_as_float(k & 0x7FFFFFFFu)
                             : __uint_as_float(~k);
}

// low 32 bits of a flat pointer into __shared__ == wave-relative LDS byte offset
__device__ __forceinline__ unsigned int lds_off(const void* p) {
    return (unsigned int)(size_t)p;
}

// ---- C = A[MxK] @ B[Kx128], fp32 accum via V_WMMA_F32_16X16X4_F32 ------------
// block = 256 threads (8 wave32). Block -> 16 rows x 128 cols. Wave w -> cols
// [16w,16w+16). K chunked by 64 through double-buffered LDS filled with
// GLOBAL_LOAD_ASYNC_TO_LDS_B128 (ASYNCcnt), overlapping next chunk's loads
// with the current chunk's 16 WMMAs.
__global__ __launch_bounds__(256) void gemm_f32_wmma(
    const float* __restrict__ A, const float* __restrict__ B,
    float* __restrict__ C, int M, int K)
{
    __shared__ float lds_a[2][16 * LDA];
    __shared__ float lds_b[2][KC * LDB];

    const int t    = threadIdx.x;
    const int wave = t >> 5;
    const int lane = t & 31;
    const int half = lane >> 4;   // K-split half per ISA f32 A/B layout
    const int r    = lane & 15;
    const int row0 = blockIdx.x * 16;
    const int col0 = wave * 16;

    // per-thread staging coordinates (constant across chunks)
    const int ar  = t >> 4;            // A row        (16 rows x 16 float4/row)
    const int ac4 = (t & 15) << 2;     // A col (floats)

    auto stage = [&](int kc, int sel) {
        // A chunk: 16x64 floats = 256 float4, one async b128 per thread
        {
            const unsigned int lo = lds_off(&lds_a[sel][ar * LDA + ac4]);
            const unsigned int go =
                (unsigned int)(((size_t)(row0 + ar) * K + kc + ac4) * sizeof(float));
            asm volatile("global_load_async_to_lds_b128 %0, %1, %2"
                         :: "v"(lo), "v"(go), "s"(A) : "memory");
        }
        // B chunk: 64x128 floats = 2048 float4, eight async b128 per thread
        #pragma unroll
        for (int i = 0; i < 8; ++i) {
            const int j   = t + (i << 8);
            const int br  = j >> 5;
            const int bc4 = (j & 31) << 2;
            const unsigned int lo = lds_off(&lds_b[sel][br * LDB + bc4]);
            const unsigned int go =
                (unsigned int)(((size_t)(kc + br) * DCOLS + bc4) * sizeof(float));
            asm volatile("global_load_async_to_lds_b128 %0, %1, %2"
                         :: "v"(lo), "v"(go), "s"(B) : "memory");
        }
    };

    v8f acc = {};
    const int nch = K / KC;

    stage(0, 0);
    asm volatile("s_wait_asynccnt 0x0" ::: "memory");
    __syncthreads();

    for (int i = 0; i < nch; ++i) {
        const int sel = i & 1;
        if (i + 1 < nch) stage((i + 1) * KC, (i + 1) & 1);  // overlap with compute

        const float* la = &lds_a[sel][0];
        const float* lb = &lds_b[sel][0];
        #pragma unroll
        for (int k4 = 0; k4 < KC / 4; ++k4) {
            const int kb = (k4 << 2) + (half << 1);   // K = kb, kb+1 in this half
            v2f a, b;
            a.x = la[r * LDA + kb];
            a.y = la[r * LDA + kb + 1];
            b.x = lb[kb * LDB + col0 + r];
            b.y = lb[(kb + 1) * LDB + col0 + r];
            acc = __builtin_amdgcn_wmma_f32_16x16x4_f32(
                false, a, false, b, (short)0, acc, false, false);
        }

        asm volatile("s_wait_asynccnt 0x0" ::: "memory");  // next chunk landed
        __syncthreads();                                   // everyone done reading
    }

    // C/D layout: VGPR i -> row = i + 8*half, col = col0 + r
    float* Cp = C + (size_t)row0 * DCOLS + col0 + r;
    #pragma unroll
    for (int i = 0; i < 8; ++i) {
        Cp[(size_t)(i + half * 8) * DCOLS] = acc[i];
    }
}

// ---- init output accumulator with bias (+ optional MMSE embedding),
//      and reset per-node max-key / denominator ------------------------------
__global__ void gat_init(
    float* __restrict__ outbuf, const float* __restrict__ bias,
    const float* __restrict__ mW, const float* __restrict__ mb,
    const float* __restrict__ mscore, int withM, int N,
    unsigned int* __restrict__ nmax, float* __restrict__ nsum)
{
    const int id = blockIdx.x * blockDim.x + threadIdx.x;
    const int total = N * DCOLS;
    if (id < total) {
        const int c = id & (DCOLS - 1);
        float v = bias[c];
        if (withM) v += mscore[0] * mW[c] + mb[c];
        outbuf[id] = v;
    }
    if (id < N) {
        nmax[id] = 0u;      // key-space minimum (below any real float key)
        nsum[id] = 0.f;
    }
}

// ---- e = att . leakyrelu(XL[src] + XR[dst]); segment max over dst -----------
// one wave32 per edge; edges >= E are self loops (src = dst = id - E)
__global__ __launch_bounds__(256) void gat_edge_logits(
    const float* __restrict__ XL, const float* __restrict__ XR,
    const float* __restrict__ att, const int* __restrict__ ei,
    int E, int EA, float* __restrict__ eout, unsigned int* __restrict__ nmax)
{
    const int lane = threadIdx.x & 31;
    const int edge = blockIdx.x * 8 + (threadIdx.x >> 5);
    if (edge >= EA) return;
    int s, d;
    if (edge < E) { s = ei[edge]; d = ei[E + edge]; }
    else          { s = edge - E; d = s; }

    const float* xl = XL + (size_t)s * DCOLS;
    const float* xr = XR + (size_t)d * DCOLS;
    float part = 0.f;
    #pragma unroll
    for (int j = 0; j < 4; ++j) {
        const int c = lane + (j << 5);
        float h = xl[c] + xr[c];
        h = (h > 0.f) ? h : 0.2f * h;          // LeakyReLU(0.2)
        part += h * att[c];
    }
    #pragma unroll
    for (int off = 16; off > 0; off >>= 1)
        part += __shfl_xor(part, off, 32);

    if (lane == 0) {
        eout[edge] = part;
        atomicMax(&nmax[d], fkey(part));
    }
}

// ---- e_exp = exp(e - max[dst]); segment sum over dst ------------------------
__global__ void gat_edge_expsum(
    float* __restrict__ ebuf, const int* __restrict__ ei,
    int E, int EA, const unsigned int* __restrict__ nmax,
    float* __restrict__ nsum)
{
    const int id = blockIdx.x * blockDim.x + threadIdx.x;
    if (id >= EA) return;
    const int d = (id < E) ? ei[E + id] : (id - E);
    const float m  = funkey(nmax[d]);
    const float ex = expf(ebuf[id] - m);
    ebuf[id] = ex;
    atomicAdd(&nsum[d], ex);
}

// ---- alpha = e_exp / denom[dst]; out[dst] += alpha * XL[src] ----------------
// one wave32 per edge; alpha_out may alias ebuf (read-before-write per wave)
__global__ __launch_bounds__(256) void gat_edge_scatter(
    const float* __restrict__ ebuf, const float* __restrict__ nsum,
    const int* __restrict__ ei, const float* __restrict__ XL,
    int E, int EA, float* __restrict__ alpha_out, float* __restrict__ outbuf)
{
    const int lane = threadIdx.x & 31;
    const int edge = blockIdx.x * 8 + (threadIdx.x >> 5);
    if (edge >= EA) return;
    int s, d;
    if (edge < E) { s = ei[edge]; d = ei[E + edge]; }
    else          { s = edge - E; d = s; }

    const float alpha = ebuf[edge] / nsum[d];
    if (lane == 0) alpha_out[edge] = alpha;

    const float* xl = XL + (size_t)s * DCOLS;
    float*       ob = outbuf + (size_t)d * DCOLS;
    #pragma unroll
    for (int j = 0; j < 4; ++j) {
        const int c = lane + (j << 5);
        atomicAdd(&ob[c], xl[c] * alpha);
    }
}

extern "C" void kernel_launch(void* const* d_in, const int* in_sizes, int n_in,
                              void* d_out, int out_size, void* d_ws, size_t ws_size,
                              hipStream_t stream)
{
    const float* x     = (const float*)d_in[0];
    const int*   ei    = (const int*)  d_in[1];
    const float* mmse  = (const float*)d_in[2];
    const float* Wl1   = (const float*)d_in[3];
    const float* Wr1   = (const float*)d_in[4];
    const float* att1  = (const float*)d_in[5];
    const float* b1    = (const float*)d_in[6];
    const float* Wl2   = (const float*)d_in[7];
    const float* Wr2   = (const float*)d_in[8];
    const float* att2  = (const float*)d_in[9];
    const float* b2    = (const float*)d_in[10];
    const float* mW    = (const float*)d_in[11];
    const float* mb    = (const float*)d_in[12];

    const int T  = in_sizes[3] / DCOLS;   // 512
    const int N  = in_sizes[0] / T;       // 50000
    const int E  = in_sizes[1] / 2;       // 800000
    const int EA = E + N;                 // + self loops

    // workspace layout
    float*        XL   = (float*)d_ws;
    float*        XR   = XL + (size_t)N * DCOLS;
    float*        H1   = XR + (size_t)N * DCOLS;
    float*        ebuf = H1 + (size_t)N * DCOLS;
    unsigned int* nmax = (unsigned int*)(ebuf + EA);
    float*        nsum = (float*)(nmax + N);

    float* gout   = (float*)d_out;                 // graph_features [N,128]
    float* alpha1 = gout + (size_t)N * DCOLS;      // attn_weights   [EA]

    const int tiles      = N / 16;                 // 50000 % 16 == 0
    const int initBlocks = (N * DCOLS + 255) / 256;
    const int waveBlocks = (EA + 7) / 8;
    const int edgeBlocks = (EA + 255) / 256;

    // ---------------- layer 1 ----------------
    gat_init<<<initBlocks, 256, 0, stream>>>(H1, b1, mW, mb, mmse, 0, N, nmax, nsum);
    gemm_f32_wmma<<<tiles, 256, 0, stream>>>(x, Wl1, XL, N, T);
    gemm_f32_wmma<<<tiles, 256, 0, stream>>>(x, Wr1, XR, N, T);
    gat_edge_logits<<<waveBlocks, 256, 0, stream>>>(XL, XR, att1, ei, E, EA, ebuf, nmax);
    gat_edge_expsum<<<edgeBlocks, 256, 0, stream>>>(ebuf, ei, E, EA, nmax, nsum);
    gat_edge_scatter<<<waveBlocks, 256, 0, stream>>>(ebuf, nsum, ei, XL, E, EA, alpha1, H1);

    // ---------------- layer 2 ----------------
    gat_init<<<initBlocks, 256, 0, stream>>>(gout, b2, mW, mb, mmse, 1, N, nmax, nsum);
    gemm_f32_wmma<<<tiles, 256, 0, stream>>>(H1, Wl2, XL, N, DCOLS);
    gemm_f32_wmma<<<tiles, 256, 0, stream>>>(H1, Wr2, XR, N, DCOLS);
    gat_edge_logits<<<waveBlocks, 256, 0, stream>>>(XL, XR, att2, ei, E, EA, ebuf, nmax);
    gat_edge_expsum<<<edgeBlocks, 256, 0, stream>>>(ebuf, ei, E, EA, nmax, nsum);
    // layer-2 alpha is discarded; alias alpha_out onto ebuf (safe: per-wave RAW order)
    gat_edge_scatter<<<waveBlocks, 256, 0, stream>>>(ebuf, nsum, ei, XL, E, EA, ebuf, gout);
}